// GCN_20091857011301
// MI455X (gfx1250) — compile-verified
//
#include <hip/hip_runtime.h>
#include <hip/hip_bf16.h>
#include <math.h>

typedef __attribute__((ext_vector_type(2))) float v2f;
typedef __attribute__((ext_vector_type(8))) float v8f;

__device__ __forceinline__ float relu_f(float v) { return v > 0.f ? v : 0.f; }

// ---------------------------------------------------------------------------
// Kernel 0: zero the atomic-accumulator region of the workspace (s + agg2).
// Must run every launch: the harness does not re-poison/zero between replays.
// ---------------------------------------------------------------------------
__global__ void k_zero(float4* __restrict__ p, long n4) {
    long i = (long)blockIdx.x * blockDim.x + threadIdx.x;
    long stride = (long)gridDim.x * blockDim.x;
    float4 z; z.x = 0.f; z.y = 0.f; z.z = 0.f; z.w = 0.f;
    for (; i < n4; i += stride) p[i] = z;
}

// ---------------------------------------------------------------------------
// Kernel 1: layer-1 scalar edge scatter. Since C_in == 1, (x@W1)[src]*w
// factors as (x[src]*w) (scalar) times the W1 row, so we only scatter the
// scalar: s[dst] += x[src]*w.  W1/b1 are applied later in-register.
// ---------------------------------------------------------------------------
__global__ void k_scatter1(const float* __restrict__ x,
                           const int* __restrict__ src,
                           const int* __restrict__ dst,
                           const float* __restrict__ w,
                           float* __restrict__ s, int E) {
    int e = blockIdx.x * blockDim.x + threadIdx.x;
    if (e < E) {
        float v = x[src[e]] * w[e];
        unsafeAtomicAdd(&s[dst[e]], v);   // global_atomic_add_f32
    }
}

// ---------------------------------------------------------------------------
// Kernel 2: fused layer-1 feature construction + layer-2 GEMM via WMMA.
// One wave32 handles a 16-node tile: D[16x32] = relu(s*W1+b1)[16x64] @ W2[64x32]
// using V_WMMA_F32_16X16X4_F32, K-chunked 16x (K=4 each), two N-halves.
// A fragment layout (32-bit A 16x4): lane L<16 -> M=L, holds K={k0,k0+1};
// lane L>=16 -> M=L-16, holds K={k0+2,k0+3}.  A is built in-register, so the
// [N x 64] h1 intermediate (358 MB) never touches HBM.
// ---------------------------------------------------------------------------
#define WAVES_PER_BLOCK 4
__global__ void k_layer1_gemm(const float* __restrict__ s,
                              const float* __restrict__ W1,
                              const float* __restrict__ b1,
                              const float* __restrict__ W2,
                              float* __restrict__ h2pre,
                              int Nn) {
    const int lane  = threadIdx.x & 31;
    const int wave  = threadIdx.x >> 5;
    const int tiles = Nn >> 4;                      // N % 16 == 0 (1.4M)
    const int tile  = blockIdx.x * WAVES_PER_BLOCK + wave;
    if (tile >= tiles) return;                      // whole-wave uniform exit

    const int node0 = tile << 4;
    const int row   = lane & 15;                    // M index for A, N index for B/D
    const int khalf = lane >> 4;                    // 0: K={k0,k0+1}, 1: K={k0+2,k0+3}
    const int col   = row;

    const float sv = s[node0 + row];

    v8f c0 = {};   // out columns 0..15
    v8f c1 = {};   // out columns 16..31
#pragma unroll
    for (int kk = 0; kk < 16; ++kk) {
        const int k0 = kk * 4 + khalf * 2;
        // A = relu(s * W1 + b1), built in-register
        v2f a;
        a.x = relu_f(fmaf(sv, W1[k0],     b1[k0]));
        a.y = relu_f(fmaf(sv, W1[k0 + 1], b1[k0 + 1]));
        // B fragments from W2[64x32] row-major (8 KB, stays in WGP$/L2)
        v2f bA, bB;
        bA.x = W2[k0 * 32 + col];
        bA.y = W2[(k0 + 1) * 32 + col];
        bB.x = W2[k0 * 32 + 16 + col];
        bB.y = W2[(k0 + 1) * 32 + 16 + col];
        c0 = __builtin_amdgcn_wmma_f32_16x16x4_f32(false, a, false, bA,
                                                   (short)0, c0, false, false);
        c1 = __builtin_amdgcn_wmma_f32_16x16x4_f32(false, a, false, bB,
                                                   (short)0, c1, false, false);
    }
    // D layout (32-bit C/D 16x16): VGPR j, lanes 0-15 -> M=j, lanes 16-31 -> M=j+8
#pragma unroll
    for (int j = 0; j < 8; ++j) {
        const int m = j + khalf * 8;
        float* o = h2pre + (long)(node0 + m) * 32;
        o[col]      = c0[j];
        o[16 + col] = c1[j];
    }
}

// ---------------------------------------------------------------------------
// Kernel 3: layer-2 edge scatter. One wave per edge, lane == channel:
// coalesced 128B gather of h2pre[src], scaled, f32 atomics into agg2[dst].
// ---------------------------------------------------------------------------
__global__ void k_scatter2(const float* __restrict__ h2pre,
                           const int* __restrict__ src,
                           const int* __restrict__ dst,
                           const float* __restrict__ w,
                           float* __restrict__ agg2, int E) {
    int e = blockIdx.x * (blockDim.x >> 5) + (threadIdx.x >> 5);
    int c = threadIdx.x & 31;
    if (e < E) {
        int sN = src[e], dN = dst[e];
        float wv = w[e];
        float v = h2pre[(long)sN * 32 + c] * wv;
        unsafeAtomicAdd(&agg2[(long)dN * 32 + c], v);
    }
}

// ---------------------------------------------------------------------------
// Kernel 4a: h2 = relu(agg2 + b2); per-node mean over the 32 channels.
// ---------------------------------------------------------------------------
__global__ void k_node_mean(const float* __restrict__ agg2,
                            const float* __restrict__ b2,
                            float* __restrict__ m, int Nn) {
    int n = blockIdx.x * blockDim.x + threadIdx.x;
    if (n < Nn) {
        const float4* a  = (const float4*)(agg2 + (long)n * 32);
        const float4* bb = (const float4*)b2;
        float sum = 0.f;
#pragma unroll
        for (int j = 0; j < 8; ++j) {
            float4 v = a[j];
            float4 b = bb[j];
            sum += relu_f(v.x + b.x) + relu_f(v.y + b.y) +
                   relu_f(v.z + b.z) + relu_f(v.w + b.w);
        }
        m[n] = sum * (1.f / 32.f);
    }
}

// ---------------------------------------------------------------------------
// Kernel 4b: per-graph 14-tap linear + sigmoid.
// ---------------------------------------------------------------------------
__global__ void k_graph_out(const float* __restrict__ m,
                            const float* __restrict__ Wl,
                            const float* __restrict__ bl,
                            float* __restrict__ out, int G) {
    int g = blockIdx.x * blockDim.x + threadIdx.x;
    if (g < G) {
        float acc = bl[0];
#pragma unroll
        for (int k = 0; k < 14; ++k) acc = fmaf(m[g * 14 + k], Wl[k], acc);
        out[g] = 1.f / (1.f + __expf(-acc));
    }
}

// ---------------------------------------------------------------------------
extern "C" void kernel_launch(void* const* d_in, const int* in_sizes, int n_in,
                              void* d_out, int out_size, void* d_ws, size_t ws_size,
                              hipStream_t stream) {
    const float* x  = (const float*)d_in[0];
    const int*   ei = (const int*)d_in[1];
    const float* ew = (const float*)d_in[2];
    const float* W1 = (const float*)d_in[3];
    const float* b1 = (const float*)d_in[4];
    const float* W2 = (const float*)d_in[5];
    const float* b2 = (const float*)d_in[6];
    const float* Wl = (const float*)d_in[7];
    const float* bl = (const float*)d_in[8];
    float* out = (float*)d_out;

    const int Nn = in_sizes[0];     // 1,400,000 nodes (x is [N,1])
    const int E  = in_sizes[2];     // 2,600,000 edges
    const int G  = out_size;        // 100,000 graphs

    const int* src = ei;            // edge_index[0, :]
    const int* dst = ei + E;        // edge_index[1, :]

    // Workspace layout (floats): [ s : N ][ agg2 : 32N ][ h2pre : 32N ]
    // s + agg2 are contiguous so one zero pass covers both atomic targets.
    float* s     = (float*)d_ws;
    float* agg2  = s + Nn;
    float* h2pre = agg2 + (long)Nn * 32;
    float* m     = s;               // s is dead after k_layer1_gemm; reuse for means

    long n4 = ((long)Nn * 33 + 3) / 4;
    k_zero<<<2048, 256, 0, stream>>>((float4*)d_ws, n4);

    k_scatter1<<<(E + 255) / 256, 256, 0, stream>>>(x, src, dst, ew, s, E);

    int tiles   = Nn >> 4;
    int blocks2 = (tiles + WAVES_PER_BLOCK - 1) / WAVES_PER_BLOCK;
    k_layer1_gemm<<<blocks2, 32 * WAVES_PER_BLOCK, 0, stream>>>(s, W1, b1, W2, h2pre, Nn);

    const int epb = 256 / 32;       // edges per 256-thread block (wave per edge)
    k_scatter2<<<(E + epb - 1) / epb, 256, 0, stream>>>(h2pre, src, dst, ew, agg2, E);

    k_node_mean<<<(Nn + 255) / 256, 256, 0, stream>>>(agg2, b2, m, Nn);
    k_graph_out<<<(G + 255) / 256, 256, 0, stream>>>(m, Wl, bl, out, G);
}